// CrossEntropy_SNNL_52003464020705
// MI455X (gfx1250) — compile-verified
//
#include <hip/hip_runtime.h>
#include <hip/hip_bf16.h>
#include <math.h>

// ---------------------------------------------------------------------------
// CE + SNNL fused pipeline for MI455X (gfx1250, wave32, WMMA)
//   d_in[0] = x_r   [8192 x 256]  f32
//   d_in[1] = y_    [8192 x 1000] f32 (logits)
//   d_in[2] = y     [8192]        int labels
//   d_out[0]        scalar f32  = CE + 0.1 * SNNL
// ---------------------------------------------------------------------------

#define BROWS 8192
#define DK    256
#define NCLS  1000

typedef __attribute__((ext_vector_type(2))) float v2f;
typedef __attribute__((ext_vector_type(8))) float v8f;

// ---------------------------------------------------------------------------
// Zero the per-row accumulators and the CE accumulator.
// ---------------------------------------------------------------------------
__global__ void __launch_bounds__(256) init_accum(float* __restrict__ top,
                                                  float* __restrict__ bot,
                                                  float* __restrict__ ce) {
    int i = blockIdx.x * blockDim.x + threadIdx.x;
    if (i < BROWS) { top[i] = 0.0f; bot[i] = 0.0f; }
    if (i == 0)    { ce[0] = 0.0f; }
}

// ---------------------------------------------------------------------------
// Row L2-normalize: xn = x / max(||x||, 1e-8).  One block (256 thr) per row.
// ---------------------------------------------------------------------------
__global__ void __launch_bounds__(256) norm_rows(const float* __restrict__ x,
                                                 float* __restrict__ xn) {
    const int row = blockIdx.x;
    const int t   = threadIdx.x;           // 0..255, one element each (DK==256)
    float v  = x[(size_t)row * DK + t];
    float ss = v * v;
    #pragma unroll
    for (int m = 16; m >= 1; m >>= 1) ss += __shfl_xor(ss, m, 32);
    __shared__ float sh[8];
    if ((t & 31) == 0) sh[t >> 5] = ss;
    __syncthreads();
    float tot = 0.0f;
    #pragma unroll
    for (int w = 0; w < 8; ++w) tot += sh[w];
    const float nrm = fmaxf(sqrtf(tot), 1e-8f);
    xn[(size_t)row * DK + t] = v / nrm;
}

// ---------------------------------------------------------------------------
// Fused sim-GEMM + SNNL epilogue.
//   grid = (64, 512): blockIdx.y = M-tile (16 rows), blockIdx.x = group of
//   8 N-tiles (128 cols).  8 waves/block, one 16x16 WMMA tile per wave,
//   K=256 via 64 x v_wmma_f32_16x16x4_f32.
//   Epilogue: E = exp((sim-1)/Tp), mask diagonal, split by class match,
//   reduce over N within the wave, across waves in LDS, one atomic per row.
// ---------------------------------------------------------------------------
__global__ void __launch_bounds__(256) snnl_gemm(const float* __restrict__ xn,
                                                 const int* __restrict__ y,
                                                 float* __restrict__ top,
                                                 float* __restrict__ bot) {
    const float invTp = 1.0f / (0.5f + 1e-6f);

    const int tm   = blockIdx.y;                 // 0..511  (M tile)
    const int wave = threadIdx.x >> 5;           // 0..7
    const int lane = threadIdx.x & 31;
    const int lr   = lane & 15;
    const int half = lane >> 4;
    const int tn   = blockIdx.x * 8 + wave;      // 0..511  (N tile)

    // fp32 WMMA 16x16x4 operand layout (ISA 7.12.2):
    //   A[m][k]: lane = m + 16*(k>>1), vgpr = k&1  -> float2 load per lane
    //   B[k][n]: lane = n + 16*(k>>1), vgpr = k&1  (B = xn^T, same pattern)
    const int row_a = tm * 16 + lr;
    const int row_b = tn * 16 + lr;
    const int koff  = half * 2;
    const float* pa = xn + (size_t)row_a * DK + koff;
    const float* pb = xn + (size_t)row_b * DK + koff;

    v8f acc = {};
    #pragma unroll 8
    for (int k = 0; k < DK; k += 4) {
        v2f a = *reinterpret_cast<const v2f*>(pa + k);
        v2f b = *reinterpret_cast<const v2f*>(pb + k);
        acc = __builtin_amdgcn_wmma_f32_16x16x4_f32(
            /*neg_a=*/false, a, /*neg_b=*/false, b,
            /*c_mod=*/(short)0, acc, /*reuse_a=*/false, /*reuse_b=*/false);
    }

    // ---- epilogue: all 8 waves share the same 16 rows (same tm) ----
    __shared__ float sh_top[16];
    __shared__ float sh_bot[16];
    if (threadIdx.x < 16) { sh_top[threadIdx.x] = 0.0f; sh_bot[threadIdx.x] = 0.0f; }
    __syncthreads();

    const int jcol = tn * 16 + lr;   // column index owned by this lane
    const int ycol = y[jcol];

    #pragma unroll
    for (int r = 0; r < 8; ++r) {
        // C/D layout: VGPR r -> M = r (lanes 0-15), M = r+8 (lanes 16-31); N = lane&15
        const int i  = tm * 16 + r + half * 8;
        float e = expf((acc[r] - 1.0f) * invTp);
        if (i == jcol) e = 0.0f;                 // not_self mask
        float tc = (y[i] == ycol) ? e : 0.0f;    // same-class (top) term
        float bc = e;                            // all-pairs (bottom) term
        // reduce over the 16 lanes of this half (sum over N in tile)
        #pragma unroll
        for (int m = 8; m >= 1; m >>= 1) {
            tc += __shfl_xor(tc, m, 32);
            bc += __shfl_xor(bc, m, 32);
        }
        if (lr == 0) {
            atomicAdd(&sh_top[r + half * 8], tc);
            atomicAdd(&sh_bot[r + half * 8], bc);
        }
    }
    __syncthreads();
    if (threadIdx.x < 16) {
        atomicAdd(&top[tm * 16 + threadIdx.x], sh_top[threadIdx.x]);
        atomicAdd(&bot[tm * 16 + threadIdx.x], sh_bot[threadIdx.x]);
    }
}

// ---------------------------------------------------------------------------
// Cross entropy: one block (256 thr) per row; 4 logits/thread held in
// registers; block max + block sum-exp; thread 0 adds -(logit_t - max - logS).
// ---------------------------------------------------------------------------
__global__ void __launch_bounds__(256) ce_rows(const float* __restrict__ logits,
                                               const int* __restrict__ y,
                                               float* __restrict__ ce_accum) {
    const int row = blockIdx.x;
    const float* lrow = logits + (size_t)row * NCLS;
    const int t = threadIdx.x;

    float vals[4];
    float vmax = -3.0e38f;
    #pragma unroll
    for (int u = 0; u < 4; ++u) {
        const int c = t + u * 256;
        vals[u] = (c < NCLS) ? lrow[c] : -3.0e38f;
        vmax = fmaxf(vmax, vals[u]);
    }

    __shared__ float sh[8];
    #pragma unroll
    for (int m = 16; m >= 1; m >>= 1) vmax = fmaxf(vmax, __shfl_xor(vmax, m, 32));
    if ((t & 31) == 0) sh[t >> 5] = vmax;
    __syncthreads();
    float rmax = sh[0];
    #pragma unroll
    for (int w = 1; w < 8; ++w) rmax = fmaxf(rmax, sh[w]);
    __syncthreads();

    float se = 0.0f;
    #pragma unroll
    for (int u = 0; u < 4; ++u) {
        const int c = t + u * 256;
        if (c < NCLS) se += expf(vals[u] - rmax);
    }
    #pragma unroll
    for (int m = 16; m >= 1; m >>= 1) se += __shfl_xor(se, m, 32);
    if ((t & 31) == 0) sh[t >> 5] = se;
    __syncthreads();

    if (t == 0) {
        float tot = 0.0f;
        #pragma unroll
        for (int w = 0; w < 8; ++w) tot += sh[w];
        const float lt = lrow[y[row]];
        atomicAdd(ce_accum, -(lt - rmax - logf(tot)));
    }
}

// ---------------------------------------------------------------------------
// Final combine: snnl = -mean(log(top'/bot)), top'=1e-6 when no same-class
// neighbor (top stays exactly 0 since every E term is > exp(-4) > 0).
// ---------------------------------------------------------------------------
__global__ void __launch_bounds__(256) finalize_kernel(const float* __restrict__ top,
                                                       const float* __restrict__ bot,
                                                       const float* __restrict__ ce_accum,
                                                       float* __restrict__ out) {
    const int t = threadIdx.x;
    float lsum = 0.0f;
    for (int i = t; i < BROWS; i += 256) {
        float tp = top[i];
        tp = (tp > 0.0f) ? tp : 1e-6f;
        lsum += logf(tp / bot[i]);
    }
    __shared__ float sh[8];
    #pragma unroll
    for (int m = 16; m >= 1; m >>= 1) lsum += __shfl_xor(lsum, m, 32);
    if ((t & 31) == 0) sh[t >> 5] = lsum;
    __syncthreads();
    if (t == 0) {
        float s = 0.0f;
        #pragma unroll
        for (int w = 0; w < 8; ++w) s += sh[w];
        const float snnl = -(s / (float)BROWS);
        out[0] = ce_accum[0] / (float)BROWS + 0.1f * snnl;
    }
}

// ---------------------------------------------------------------------------
extern "C" void kernel_launch(void* const* d_in, const int* in_sizes, int n_in,
                              void* d_out, int out_size, void* d_ws, size_t ws_size,
                              hipStream_t stream) {
    const float* x_r      = (const float*)d_in[0];   // [8192,256]
    const float* y_logits = (const float*)d_in[1];   // [8192,1000]
    const int*   y        = (const int*)d_in[2];     // [8192]
    float* out = (float*)d_out;

    // workspace layout
    char*  ws  = (char*)d_ws;
    float* xn  = (float*)ws;                                    // 8 MB
    float* top = (float*)(ws + (size_t)BROWS * DK * sizeof(float));
    float* bot = top + BROWS;
    float* ce  = bot + BROWS;

    init_accum<<<(BROWS + 255) / 256, 256, 0, stream>>>(top, bot, ce);
    norm_rows <<<BROWS, 256, 0, stream>>>(x_r, xn);

    dim3 grid(BROWS / (16 * 8), BROWS / 16);   // (64, 512)
    snnl_gemm <<<grid, 256, 0, stream>>>(xn, y, top, bot);

    ce_rows   <<<BROWS, 256, 0, stream>>>(y_logits, y, ce);
    finalize_kernel<<<1, 256, 0, stream>>>(top, bot, ce, out);
}